// GCN_78417512890502
// MI455X (gfx1250) — compile-verified
//
#include <hip/hip_runtime.h>

// ---------------------------------------------------------------------------
// 3-layer GCN on gfx1250 (MI455X), wave32 — atomic-free CSR-gather design.
//
//   CSR build : int histogram -> single-block exclusive scan -> cursor fill
//               (only 2*E int atomics; no float atomics anywhere)
//   L1        : input dim == 1 -> scalar gather s[i], h1 = relu(s * W1 + b1)
//   L2 GEMM   : t2 = h1 @ W2 with V_WMMA_F32_16X16X4_F32 (exact fp32);
//               W2 staged to LDS with GLOBAL_LOAD_ASYNC_TO_LDS_B128
//               (gfx1250 async copy engine, ASYNCcnt-tracked)
//   L2+L3     : fused per-node gather of t2 rows (51 MB -> L2-resident in the
//               192 MB L2) + relu(.+b2).W3 in-wave reduction -> t3; the agg
//               matrix never touches memory (saves ~100 MB of traffic)
//   L3 agg    : scalar CSR gather; log_softmax over 1 column == z - z
//
// Roofline: ~740k row-gathers x 512 B from L2 (~0.38 GB) + 102 MB GEMM
// traffic -> tens of microseconds at 23.3 TB/s HBM / multi-TB/s L2.
// ---------------------------------------------------------------------------

typedef float v2f __attribute__((ext_vector_type(2)));
typedef float v8f __attribute__((ext_vector_type(8)));
typedef __attribute__((address_space(3))) float lds_f32;

#define HID 128

// ---- CSR build -------------------------------------------------------------
__global__ void k_zero_int(int* p, int n) {
  int i = blockIdx.x * blockDim.x + threadIdx.x;
  if (i < n) p[i] = 0;
}

__global__ void k_count(const int* __restrict__ dst, int* __restrict__ cnt, int e) {
  int i = blockIdx.x * blockDim.x + threadIdx.x;
  if (i < e) atomicAdd(&cnt[dst[i]], 1);
}

// Single-block exclusive scan (1024 threads = 32 waves), rowStart[n] = total.
__global__ __launch_bounds__(1024)
void k_scan(const int* __restrict__ cnt, int* __restrict__ rowStart, int n) {
  __shared__ int waveSums[32];
  __shared__ int running;
  const int lane = threadIdx.x & 31;
  const int w    = threadIdx.x >> 5;
  if (threadIdx.x == 0) running = 0;
  __syncthreads();
  for (int base = 0; base < n; base += 1024) {
    int i = base + (int)threadIdx.x;
    int v = (i < n) ? cnt[i] : 0;
    int sv = v;  // inclusive scan within wave
#pragma unroll
    for (int off = 1; off < 32; off <<= 1) {
      int t = __shfl_up(sv, off, 32);
      if (lane >= off) sv += t;
    }
    if (lane == 31) waveSums[w] = sv;
    __syncthreads();
    if (w == 0) {  // scan the 32 wave sums
      int ws = waveSums[lane];
#pragma unroll
      for (int off = 1; off < 32; off <<= 1) {
        int t = __shfl_up(ws, off, 32);
        if (lane >= off) ws += t;
      }
      waveSums[lane] = ws;
    }
    __syncthreads();
    int waveOff = (w == 0) ? 0 : waveSums[w - 1];
    if (i < n) rowStart[i] = running + waveOff + sv - v;  // exclusive
    __syncthreads();
    if (threadIdx.x == 0) running += waveSums[31];
    __syncthreads();
  }
  if (threadIdx.x == 0) rowStart[n] = running;
}

// cursor = rowStart copy; dinv = rsqrt(indeg + 1)  (self-loop)
__global__ void k_prep(const int* __restrict__ rowStart, const int* __restrict__ cnt,
                       int* __restrict__ cursor, float* __restrict__ dinv, int n) {
  int i = blockIdx.x * blockDim.x + threadIdx.x;
  if (i < n) {
    cursor[i] = rowStart[i];
    dinv[i] = rsqrtf((float)cnt[i] + 1.0f);
  }
}

__global__ void k_fill(const int* __restrict__ src, const int* __restrict__ dst,
                       int* __restrict__ cursor, int* __restrict__ csrSrc, int e) {
  int i = blockIdx.x * blockDim.x + threadIdx.x;
  if (i < e) {
    int pos = atomicAdd(&cursor[dst[i]], 1);
    csrSrc[pos] = src[i];
  }
}

// ---- layer 1: fused scalar gather + rank-1 expansion ----------------------
__global__ void k_l1(const int* __restrict__ rowStart, const int* __restrict__ csrSrc,
                     const float* __restrict__ x, const float* __restrict__ dinv,
                     const float* __restrict__ W1, const float* __restrict__ b1,
                     float* __restrict__ h1, int n) {
  int gid = blockIdx.x * blockDim.x + threadIdx.x;
  int node = gid >> 5, lane = gid & 31;
  if (node >= n) return;
  const int e0 = rowStart[node], e1 = rowStart[node + 1];
  float p = 0.0f;
  for (int j = e0 + lane; j < e1; j += 32) {
    int sn = csrSrc[j];
    p += dinv[sn] * x[sn];
  }
#pragma unroll
  for (int off = 16; off > 0; off >>= 1) p += __shfl_xor(p, off, 32);
  const float di = dinv[node];
  const float s = di * p + di * di * x[node];  // all lanes hold s
  const int q = lane * 4;
  float4 w = *(const float4*)(W1 + q);
  float4 bb = *(const float4*)(b1 + q);
  float4 r;
  r.x = fmaxf(s * w.x + bb.x, 0.0f);
  r.y = fmaxf(s * w.y + bb.y, 0.0f);
  r.z = fmaxf(s * w.z + bb.z, 0.0f);
  r.w = fmaxf(s * w.w + bb.w, 0.0f);
  *(float4*)(h1 + (size_t)node * HID + q) = r;
}

// ---- layer 2 GEMM: t2 = h1 @ W2 via V_WMMA_F32_16X16X4_F32 ----------------
// Block = 256 threads = 8 waves, 128 rows per block. W2 staged to LDS with
// the gfx1250 async copy engine (GLOBAL_LOAD_ASYNC_TO_LDS_B128 + ASYNCcnt).
__global__ __launch_bounds__(256)
void k_gemm_h1w2(const float* __restrict__ A,   // h1  [n,128]
                 const float* __restrict__ W,   // W2  [128,128]
                 float* __restrict__ C,         // t2  [n,128]
                 int n) {
  __shared__ float Bs[HID * HID];

  // Async stage: each lane issues 16 x 16B descriptors; copy engine moves
  // 64 KB global -> LDS without touching VGPRs.
  {
    unsigned loff = (unsigned)(unsigned long long)(lds_f32*)&Bs[threadIdx.x * 4];
    unsigned long long gaddr =
        (unsigned long long)(const void*)(W + threadIdx.x * 4);
#pragma unroll
    for (int it = 0; it < 16; ++it) {
      unsigned l = loff + it * 4096;
      unsigned long long g = gaddr + it * 4096;
      asm volatile("global_load_async_to_lds_b128 %0, %1, off"
                   :: "v"(l), "v"(g)
                   : "memory");
    }
    asm volatile("s_wait_asynccnt 0x0" ::: "memory");
  }
  __syncthreads();

  const int wave = threadIdx.x >> 5;
  const int lane = threadIdx.x & 31;
  const int half = lane >> 4;
  const int l    = lane & 15;
  const int rowBase = blockIdx.x * 128 + wave * 16;

  int arow = rowBase + l;            // clamp so EXEC stays all-ones for WMMA
  if (arow >= n) arow = n - 1;
  const float* __restrict__ Arow = A + (size_t)arow * HID;

  v8f acc[8] = {};
  for (int k0 = 0; k0 < HID; k0 += 4) {
    // A 16x4 f32 fragment: v0 -> K = k0+2*half, v1 -> K = k0+2*half+1
    v2f a;
    a.x = Arow[k0 + 2 * half];
    a.y = Arow[k0 + 2 * half + 1];
#pragma unroll
    for (int t = 0; t < 8; ++t) {
      const int ncol = t * 16 + l;
      v2f b;  // B 4x16 fragment (mirrors A, transposed)
      b.x = Bs[(k0 + 2 * half) * HID + ncol];
      b.y = Bs[(k0 + 2 * half + 1) * HID + ncol];
      acc[t] = __builtin_amdgcn_wmma_f32_16x16x4_f32(
          false, a, false, b, (short)0, acc[t], false, false);
    }
  }

  // C/D layout: lane<16 -> M=v, N=lane ; lane>=16 -> M=v+8, N=lane-16
  if (rowBase + 16 <= n) {           // uniform fast path (all blocks but last)
#pragma unroll
    for (int t = 0; t < 8; ++t) {
      const int ncol = t * 16 + l;
#pragma unroll
      for (int v = 0; v < 8; ++v)
        C[(size_t)(rowBase + v + 8 * half) * HID + ncol] = acc[t][v];
    }
  } else {
#pragma unroll
    for (int t = 0; t < 8; ++t) {
      const int ncol = t * 16 + l;
#pragma unroll
      for (int v = 0; v < 8; ++v) {
        const int row = rowBase + v + 8 * half;
        if (row < n) C[(size_t)row * HID + ncol] = acc[t][v];
      }
    }
  }
}

// ---- fused layer-2 aggregation + layer-3 transform ------------------------
__global__ void k_l2l3(const int* __restrict__ rowStart, const int* __restrict__ csrSrc,
                       const float* __restrict__ t2, const float* __restrict__ dinv,
                       const float* __restrict__ b2, const float* __restrict__ W3,
                       float* __restrict__ t3, int n) {
  int gid = blockIdx.x * blockDim.x + threadIdx.x;
  int node = gid >> 5, lane = gid & 31;
  if (node >= n) return;
  const int e0 = rowStart[node], e1 = rowStart[node + 1];
  const float di = dinv[node];
  const int q = lane * 4;

  // self-loop term
  float4 v0 = *(const float4*)(t2 + (size_t)node * HID + q);
  float sn = di * di;
  float4 acc;
  acc.x = sn * v0.x; acc.y = sn * v0.y; acc.z = sn * v0.z; acc.w = sn * v0.w;

  int j = e0;
  int sc = (j < e1) ? csrSrc[j] : 0;
  while (j < e1) {
    int nj = j + 1;
    int scNext = (nj < e1) ? csrSrc[nj] : sc;
    __builtin_prefetch(t2 + (size_t)scNext * HID + q, 0, 0);  // global_prefetch_b8
    float norm = dinv[sc] * di;
    float4 v = *(const float4*)(t2 + (size_t)sc * HID + q);
    acc.x += norm * v.x; acc.y += norm * v.y;
    acc.z += norm * v.z; acc.w += norm * v.w;
    sc = scNext; j = nj;
  }

  // h2 = relu(acc + b2); partial dot with W3, reduce in-wave
  float4 bb = *(const float4*)(b2 + q);
  float4 w = *(const float4*)(W3 + q);
  float p = fmaxf(acc.x + bb.x, 0.0f) * w.x + fmaxf(acc.y + bb.y, 0.0f) * w.y +
            fmaxf(acc.z + bb.z, 0.0f) * w.z + fmaxf(acc.w + bb.w, 0.0f) * w.w;
#pragma unroll
  for (int off = 16; off > 0; off >>= 1) p += __shfl_xor(p, off, 32);
  if (lane == 0) t3[node] = p;
}

// ---- layer 3 scalar gather + log_softmax(1 col) == 0 ----------------------
__global__ void k_l3out(const int* __restrict__ rowStart, const int* __restrict__ csrSrc,
                        const float* __restrict__ t3, const float* __restrict__ dinv,
                        const float* __restrict__ b3, float* __restrict__ out, int n) {
  int i = blockIdx.x * blockDim.x + threadIdx.x;
  if (i >= n) return;
  const int e0 = rowStart[i], e1 = rowStart[i + 1];
  const float di = dinv[i];
  float p = 0.0f;
  for (int j = e0; j < e1; ++j) {
    int sn = csrSrc[j];
    p += dinv[sn] * t3[sn];
  }
  float z = di * p + di * di * t3[i] + b3[0];
  out[i] = z - z;  // log_softmax over a single-element row
}

// ---------------------------------------------------------------------------
extern "C" void kernel_launch(void* const* d_in, const int* in_sizes, int n_in,
                              void* d_out, int out_size, void* d_ws, size_t ws_size,
                              hipStream_t stream) {
  const float* x  = (const float*)d_in[0];
  const int*   ei = (const int*)d_in[1];   // [2, E] row-major: src then dst
  const float* W1 = (const float*)d_in[2];
  const float* b1 = (const float*)d_in[3];
  const float* W2 = (const float*)d_in[4];
  const float* b2 = (const float*)d_in[5];
  const float* W3 = (const float*)d_in[6];
  const float* b3 = (const float*)d_in[7];
  float* out = (float*)d_out;

  const int N = in_sizes[0];
  const int E = in_sizes[1] / 2;
  const int* src = ei;
  const int* dst = ei + E;

  // workspace carve-out (256B-aligned chunks)
  char* ws = (char*)d_ws;
  size_t off = 0;
  auto carve = [&](size_t bytes) {
    void* p = ws + off;
    off += (bytes + 255) & ~(size_t)255;
    return p;
  };
  int*   cnt      = (int*)carve((size_t)N * 4);
  int*   rowStart = (int*)carve((size_t)(N + 1) * 4);
  int*   cursor   = (int*)carve((size_t)N * 4);
  int*   csrSrc   = (int*)carve((size_t)E * 4);
  float* dinv     = (float*)carve((size_t)N * 4);
  float* t3       = (float*)carve((size_t)N * 4);
  float* h1       = (float*)carve((size_t)N * HID * 4);
  float* t2       = (float*)carve((size_t)N * HID * 4);

  const int TB = 256;
  const int gN  = (N + TB - 1) / TB;
  const int gE  = (E + TB - 1) / TB;
  const int gN32 = (int)(((size_t)N * 32 + TB - 1) / TB);

  // CSR build + normalization
  k_zero_int<<<gN, TB, 0, stream>>>(cnt, N);
  k_count<<<gE, TB, 0, stream>>>(dst, cnt, E);
  k_scan<<<1, 1024, 0, stream>>>(cnt, rowStart, N);
  k_prep<<<gN, TB, 0, stream>>>(rowStart, cnt, cursor, dinv, N);
  k_fill<<<gE, TB, 0, stream>>>(src, dst, cursor, csrSrc, E);

  // layer 1 (fused scalar gather + rank-1 expansion)
  k_l1<<<gN32, TB, 0, stream>>>(rowStart, csrSrc, x, dinv, W1, b1, h1, N);

  // layer 2 GEMM (WMMA fp32, async-LDS staged weights)
  k_gemm_h1w2<<<(N + 127) / 128, TB, 0, stream>>>(h1, W2, t2, N);

  // fused layer-2 aggregation + layer-3 transform (atomic-free gather)
  k_l2l3<<<gN32, TB, 0, stream>>>(rowStart, csrSrc, t2, dinv, b2, W3, t3, N);

  // layer 3 scalar gather + log_softmax
  k_l3out<<<gN, TB, 0, stream>>>(rowStart, csrSrc, t3, dinv, b3, out, N);
}